// LuongAttnDecoderRNN_2130303779133
// MI455X (gfx1250) — compile-verified
//
#include <hip/hip_runtime.h>
#include <math.h>

#define B 64
#define S 4096
#define H 512
#define V 32000
#define CHUNKS 16
#define SCHUNK (S / CHUNKS)   // 256

typedef __attribute__((ext_vector_type(2))) float v2f;
typedef __attribute__((ext_vector_type(8))) float v8f;

// ---------------------------------------------------------------------------
// Generic M=64 GEMM:  out[64][N] = act( A[64][K] * W[N][K]^T + bias[N] )
// A rows optionally gathered through idx (embedding fusion).
// One wave computes a full 64-row column strip of 16 N values with
// V_WMMA_F32_16X16X4_F32. Software-pipelined: next k-step fragments are
// loaded while the 4 WMMAs of the current k-step execute. All loads use a
// uniform SGPR base + 32-bit lane offset (GVS addressing) to avoid per-lane
// 64-bit pointer math in the loop.
// ---------------------------------------------------------------------------
__global__ __launch_bounds__(256) void gemm64_wmma(
    const float* __restrict__ A, const int* __restrict__ idx,
    const float* __restrict__ W, const float* __restrict__ bias,
    float* __restrict__ out, int N, int K, int act)
{
    const int lane  = threadIdx.x & 31;
    const int wave  = threadIdx.x >> 5;
    const int n0    = blockIdx.x * 128 + wave * 16;
    const int col   = n0 + (lane & 15);
    const int kh2   = (lane >> 4) * 2;        // 0 or 2: lane's K-pair base

    // 32-bit element offsets (uniform base pointers -> saddr-form loads)
    int aoff[4];
#pragma unroll
    for (int mt = 0; mt < 4; ++mt) {
        int m = mt * 16 + (lane & 15);
        int r = idx ? idx[m] : m;
        aoff[mt] = r * K + kh2;
    }
    int woff = col * K + kh2;

    v8f acc[4] = {v8f{}, v8f{}, v8f{}, v8f{}};

    // Prologue: load k-step 0 fragments
    v2f bcur = *(const v2f*)(W + woff);
    v2f acur[4];
#pragma unroll
    for (int mt = 0; mt < 4; ++mt) acur[mt] = *(const v2f*)(A + aoff[mt]);

#pragma unroll 2
    for (int k0 = 0; k0 < K - 4; k0 += 4) {
        // Prefetch next k-step fragments (latency hidden behind 4 WMMAs)
        const int kn = k0 + 4;
        v2f bnext = *(const v2f*)(W + woff + kn);
        v2f anext[4];
#pragma unroll
        for (int mt = 0; mt < 4; ++mt)
            anext[mt] = *(const v2f*)(A + aoff[mt] + kn);

#pragma unroll
        for (int mt = 0; mt < 4; ++mt)
            acc[mt] = __builtin_amdgcn_wmma_f32_16x16x4_f32(
                false, acur[mt], false, bcur, (short)0, acc[mt], false, false);

        bcur = bnext;
#pragma unroll
        for (int mt = 0; mt < 4; ++mt) acur[mt] = anext[mt];
    }
    // Epilogue: last k-step
#pragma unroll
    for (int mt = 0; mt < 4; ++mt)
        acc[mt] = __builtin_amdgcn_wmma_f32_16x16x4_f32(
            false, acur[mt], false, bcur, (short)0, acc[mt], false, false);

    const float bv = bias[col];
#pragma unroll
    for (int mt = 0; mt < 4; ++mt) {
#pragma unroll
        for (int r = 0; r < 8; ++r) {
            int m = mt * 16 + r + 4 * kh2;    // D: M=r+8*(lane/16), N=lane%16
            float v = acc[mt][r] + bv;
            if (act) v = tanhf(v);
            out[(size_t)m * (size_t)N + col] = v;
        }
    }
}

// ---------------------------------------------------------------------------
// GRU gate fusion: h_new = (1-z)*n + z*h   (PyTorch GRU convention)
// Writes h_new to d_out and into concat staging buffer (stride 1024, col 0).
// ---------------------------------------------------------------------------
__global__ __launch_bounds__(256) void gru_gates(
    const float* __restrict__ gi, const float* __restrict__ gh,
    const float* __restrict__ h,
    float* __restrict__ hnew_out, float* __restrict__ concatA)
{
    int t = blockIdx.x * 256 + threadIdx.x;   // 0 .. B*H-1
    int b = t >> 9, j = t & (H - 1);
    size_t base = (size_t)b * (3 * H) + j;
    float ir = gi[base],         hr = gh[base];
    float iz = gi[base + H],     hz = gh[base + H];
    float in_ = gi[base + 2*H],  hn = gh[base + 2*H];
    float r = 1.f / (1.f + __expf(-(ir + hr)));
    float z = 1.f / (1.f + __expf(-(iz + hz)));
    float n = tanhf(in_ + r * hn);
    float o = (1.f - z) * n + z * h[t];
    hnew_out[t] = o;
    concatA[(size_t)b * 1024 + j] = o;
}

// ---------------------------------------------------------------------------
// Flash attention pass 1: one (batch, chunk) per block; 8 waves, each wave
// streams s = chunk*256 + 8*i + wave. Online softmax + rescaled context
// accumulation; raw energies stored for later attn_weights normalization.
// Single 512 MB read of encoder_outputs -> ~22 us at 23.3 TB/s.
// ---------------------------------------------------------------------------
__global__ __launch_bounds__(256) void attn_pass1(
    const float* __restrict__ hnew, const float* __restrict__ enc,
    float* __restrict__ energies,
    float* __restrict__ part_m, float* __restrict__ part_l,
    float* __restrict__ part_ctx)
{
    const int b     = blockIdx.x >> 4;
    const int chunk = blockIdx.x & 15;
    const int tid   = threadIdx.x;
    const int lane  = tid & 31;
    const int wave  = tid >> 5;

    __shared__ float hs[H];
    __shared__ float red_m[8], red_l[8];
    __shared__ float red_ctx[8 * H];          // 16 KB

    hs[tid]       = hnew[(size_t)b * H + tid];
    hs[tid + 256] = hnew[(size_t)b * H + tid + 256];
    __syncthreads();

    float4 hv0 = *(const float4*)&hs[lane * 16 + 0];
    float4 hv1 = *(const float4*)&hs[lane * 16 + 4];
    float4 hv2 = *(const float4*)&hs[lane * 16 + 8];
    float4 hv3 = *(const float4*)&hs[lane * 16 + 12];

    float m = -INFINITY, l = 0.f;
    float4 c0 = {0,0,0,0}, c1 = {0,0,0,0}, c2 = {0,0,0,0}, c3 = {0,0,0,0};

    const int s0 = chunk * SCHUNK;
    for (int i = 0; i < SCHUNK / 8; ++i) {
        const int s = s0 + i * 8 + wave;
        const float* ep = enc + ((size_t)s * B + b) * H + lane * 16;
        __builtin_prefetch(ep + (size_t)8 * B * H, 0, 0);  // next s for this wave
        float4 e0 = *(const float4*)(ep + 0);
        float4 e1 = *(const float4*)(ep + 4);
        float4 e2 = *(const float4*)(ep + 8);
        float4 e3 = *(const float4*)(ep + 12);

        float d = e0.x*hv0.x + e0.y*hv0.y + e0.z*hv0.z + e0.w*hv0.w
                + e1.x*hv1.x + e1.y*hv1.y + e1.z*hv1.z + e1.w*hv1.w
                + e2.x*hv2.x + e2.y*hv2.y + e2.z*hv2.z + e2.w*hv2.w
                + e3.x*hv3.x + e3.y*hv3.y + e3.z*hv3.z + e3.w*hv3.w;
#pragma unroll
        for (int o = 16; o; o >>= 1) d += __shfl_xor(d, o, 32);

        if (lane == 0) energies[(size_t)b * S + s] = d;

        float mn    = fmaxf(m, d);
        float scale = __expf(m - mn);         // first iter: exp(-inf)=0
        float p     = __expf(d - mn);
        l = l * scale + p;
        c0.x = c0.x*scale + p*e0.x; c0.y = c0.y*scale + p*e0.y;
        c0.z = c0.z*scale + p*e0.z; c0.w = c0.w*scale + p*e0.w;
        c1.x = c1.x*scale + p*e1.x; c1.y = c1.y*scale + p*e1.y;
        c1.z = c1.z*scale + p*e1.z; c1.w = c1.w*scale + p*e1.w;
        c2.x = c2.x*scale + p*e2.x; c2.y = c2.y*scale + p*e2.y;
        c2.z = c2.z*scale + p*e2.z; c2.w = c2.w*scale + p*e2.w;
        c3.x = c3.x*scale + p*e3.x; c3.y = c3.y*scale + p*e3.y;
        c3.z = c3.z*scale + p*e3.z; c3.w = c3.w*scale + p*e3.w;
        m = mn;
    }

    if (lane == 0) { red_m[wave] = m; red_l[wave] = l; }
    *(float4*)&red_ctx[wave * H + lane * 16 + 0]  = c0;
    *(float4*)&red_ctx[wave * H + lane * 16 + 4]  = c1;
    *(float4*)&red_ctx[wave * H + lane * 16 + 8]  = c2;
    *(float4*)&red_ctx[wave * H + lane * 16 + 12] = c3;
    __syncthreads();

    float bm = red_m[0];
#pragma unroll
    for (int w = 1; w < 8; ++w) bm = fmaxf(bm, red_m[w]);
    float bl = 0.f;
    float ws8[8];
#pragma unroll
    for (int w = 0; w < 8; ++w) { ws8[w] = __expf(red_m[w] - bm); bl += red_l[w] * ws8[w]; }
    if (tid == 0) {
        part_m[b * CHUNKS + chunk] = bm;
        part_l[b * CHUNKS + chunk] = bl;
    }
    for (int j = tid; j < H; j += 256) {
        float acc = 0.f;
#pragma unroll
        for (int w = 0; w < 8; ++w) acc += red_ctx[w * H + j] * ws8[w];
        part_ctx[((size_t)b * CHUNKS + chunk) * H + j] = acc;
    }
}

// ---------------------------------------------------------------------------
// Combine chunk partials -> final context; store (M, L) for attn normalize.
// ---------------------------------------------------------------------------
__global__ __launch_bounds__(256) void attn_combine(
    const float* __restrict__ part_m, const float* __restrict__ part_l,
    const float* __restrict__ part_ctx,
    float* __restrict__ context_out, float* __restrict__ concatA,
    float* __restrict__ Mout, float* __restrict__ Lout)
{
    const int b = blockIdx.x, tid = threadIdx.x;
    float M = part_m[b * CHUNKS];
#pragma unroll
    for (int i = 1; i < CHUNKS; ++i) M = fmaxf(M, part_m[b * CHUNKS + i]);
    float L = 0.f, w[CHUNKS];
#pragma unroll
    for (int i = 0; i < CHUNKS; ++i) {
        w[i] = __expf(part_m[b * CHUNKS + i] - M);
        L += part_l[b * CHUNKS + i] * w[i];
    }
    if (tid == 0) { Mout[b] = M; Lout[b] = L; }
    const float invL = 1.f / L;
    for (int j = tid; j < H; j += 256) {
        float acc = 0.f;
#pragma unroll
        for (int i = 0; i < CHUNKS; ++i)
            acc += part_ctx[((size_t)b * CHUNKS + i) * H + j] * w[i];
        float ctx = acc * invL;
        context_out[(size_t)b * H + j] = ctx;
        concatA[(size_t)b * 1024 + H + j] = ctx;   // second half of concat input
    }
}

// ---------------------------------------------------------------------------
// attn_weights[b][s] = exp(e - M[b]) / L[b]
// ---------------------------------------------------------------------------
__global__ __launch_bounds__(256) void attn_norm(
    const float* __restrict__ energies, const float* __restrict__ M,
    const float* __restrict__ L, float* __restrict__ attn_out)
{
    int t = blockIdx.x * 256 + threadIdx.x;    // 0 .. B*S-1
    int b = t >> 12;
    attn_out[t] = __expf(energies[t] - M[b]) / L[b];
}

// ---------------------------------------------------------------------------
extern "C" void kernel_launch(void* const* d_in, const int* in_sizes, int n_in,
                              void* d_out, int out_size, void* d_ws, size_t ws_size,
                              hipStream_t stream)
{
    const int*   input_seq   = (const int*)  d_in[0];
    // d_in[1] = last_context (unused by reference)
    const float* last_hidden = (const float*)d_in[2];   // (1,B,H)
    const float* enc         = (const float*)d_in[3];   // (S,B,H)
    const float* emb         = (const float*)d_in[4];   // (V,H)
    const float* W_ih        = (const float*)d_in[5];   // (3H,H)
    const float* W_hh        = (const float*)d_in[6];
    const float* b_ih        = (const float*)d_in[7];
    const float* b_hh        = (const float*)d_in[8];
    const float* W_concat    = (const float*)d_in[9];   // (H,2H)
    const float* b_concat    = (const float*)d_in[10];
    const float* W_out       = (const float*)d_in[11];  // (V,H)
    const float* b_out       = (const float*)d_in[12];

    float* out_output  = (float*)d_out;            // (B,V)
    float* out_context = out_output + (size_t)B * V;
    float* out_hnew    = out_context + (size_t)B * H;
    float* out_attn    = out_hnew + (size_t)B * H; // (B,1,S)

    float* p = (float*)d_ws;
    float* ws_gi        = p; p += (size_t)B * 3 * H;
    float* ws_gh        = p; p += (size_t)B * 3 * H;
    float* ws_concatA   = p; p += (size_t)B * 1024;
    float* ws_concatOut = p; p += (size_t)B * H;
    float* ws_energies  = p; p += (size_t)B * S;
    float* ws_pm        = p; p += (size_t)B * CHUNKS;
    float* ws_pl        = p; p += (size_t)B * CHUNKS;
    float* ws_pctx      = p; p += (size_t)B * CHUNKS * H;
    float* ws_M         = p; p += B;
    float* ws_L         = p; p += B;

    dim3 blk(256);

    // GRU input/hidden GEMMs (emb gather fused into A via idx)
    gemm64_wmma<<<12, blk, 0, stream>>>(emb, input_seq, W_ih, b_ih, ws_gi, 3 * H, H, 0);
    gemm64_wmma<<<12, blk, 0, stream>>>(last_hidden, nullptr, W_hh, b_hh, ws_gh, 3 * H, H, 0);
    gru_gates<<<(B * H) / 256, blk, 0, stream>>>(ws_gi, ws_gh, last_hidden,
                                                 out_hnew, ws_concatA);
    // Flash attention: single pass over 512 MB of encoder_outputs
    attn_pass1<<<B * CHUNKS, blk, 0, stream>>>(out_hnew, enc, ws_energies,
                                               ws_pm, ws_pl, ws_pctx);
    attn_combine<<<B, blk, 0, stream>>>(ws_pm, ws_pl, ws_pctx,
                                        out_context, ws_concatA, ws_M, ws_L);
    attn_norm<<<(B * S) / 256, blk, 0, stream>>>(ws_energies, ws_M, ws_L, out_attn);
    // concat projection with tanh (K = 2H), then output projection (N = V)
    gemm64_wmma<<<4, blk, 0, stream>>>(ws_concatA, nullptr, W_concat, b_concat,
                                       ws_concatOut, H, 1024, 1);
    gemm64_wmma<<<250, blk, 0, stream>>>(ws_concatOut, nullptr, W_out, b_out,
                                         out_output, V, H, 0);
}